// AttentionModule_25074019074516
// MI455X (gfx1250) — compile-verified
//
#include <hip/hip_runtime.h>

// Problem constants (from reference): B=128, S=2048, FEAT=512, DK=64
#define B_     128
#define S_     2048
#define FEAT_  512
#define DK_    64
#define NCHUNK 8
#define SCHUNK (S_ / NCHUNK)   // 256 rows per chunk

typedef __attribute__((ext_vector_type(2))) float v2f;
typedef __attribute__((ext_vector_type(8))) float v8f;

// ---------------------------------------------------------------------------
// Kernel 1a: q[128x64] = input_q[128x512] @ W_Q[512x64]   (WMMA f32 16x16x4)
// grid: 32 blocks of 32 threads (one wave per 16x16 output tile, 8x4 tiles)
// ---------------------------------------------------------------------------
__global__ void proj_q_kernel(const float* __restrict__ X,
                              const float* __restrict__ W,
                              float* __restrict__ Q) {
    const int mt   = blockIdx.x >> 2;      // 0..7
    const int nt   = blockIdx.x & 3;       // 0..3
    const int lane = threadIdx.x;          // 0..31 (wave32)
    const int m    = mt * 16 + (lane & 15);
    const int n    = nt * 16 + (lane & 15);
    const int kh   = (lane < 16) ? 0 : 2;  // A/B VGPR K-split per ISA layout

    v8f c = {};
    for (int k0 = 0; k0 < FEAT_; k0 += 4) {
        v2f a, b;
        a.x = X[m * FEAT_ + k0 + kh];          // A: K = kh   (vgpr0)
        a.y = X[m * FEAT_ + k0 + kh + 1];      // A: K = kh+1 (vgpr1)
        b.x = W[(k0 + kh) * DK_ + n];          // B: K = kh
        b.y = W[(k0 + kh + 1) * DK_ + n];      // B: K = kh+1
        c = __builtin_amdgcn_wmma_f32_16x16x4_f32(false, a, false, b,
                                                  (short)0, c, false, false);
    }
    const int row = mt * 16 + ((lane < 16) ? 0 : 8);
#pragma unroll
    for (int r = 0; r < 8; ++r)
        Q[(row + r) * DK_ + n] = c[r];
}

// ---------------------------------------------------------------------------
// Kernel 1b: v[128x512] = q[128x64] @ W_K^T   (W_K is [512x64], row-major)
// grid: 256 blocks of 32 threads (8 x 32 tiles of 16x16)
// ---------------------------------------------------------------------------
__global__ void proj_v_kernel(const float* __restrict__ Q,
                              const float* __restrict__ WK,
                              float* __restrict__ V) {
    const int mt   = blockIdx.x >> 5;      // 0..7
    const int nt   = blockIdx.x & 31;      // 0..31
    const int lane = threadIdx.x;
    const int m    = mt * 16 + (lane & 15);
    const int n    = nt * 16 + (lane & 15);
    const int kh   = (lane < 16) ? 0 : 2;

    v8f c = {};
    for (int k0 = 0; k0 < DK_; k0 += 4) {
        v2f a, b;
        a.x = Q[m * DK_ + k0 + kh];
        a.y = Q[m * DK_ + k0 + kh + 1];
        b.x = WK[n * DK_ + k0 + kh];       // B[k][n] = W_K[n][k]
        b.y = WK[n * DK_ + k0 + kh + 1];
        c = __builtin_amdgcn_wmma_f32_16x16x4_f32(false, a, false, b,
                                                  (short)0, c, false, false);
    }
    const int row = mt * 16 + ((lane < 16) ? 0 : 8);
#pragma unroll
    for (int r = 0; r < 8; ++r)
        V[(row + r) * FEAT_ + n] = c[r];
}

// ---------------------------------------------------------------------------
// Kernel 2: fused flash pass over one S-chunk.
// grid (NCHUNK, B), 256 threads (8 waves). Each wave owns rows
// s = chunk*256 + wave + 8*j and keeps private online-softmax state:
// running max m, running sum l, ctx accumulator (512 f32 striped as 4 float4
// per lane, f = 4*(lane + 32*i)). Single read of input_k; float4 coalesced.
// ---------------------------------------------------------------------------
__global__ void attn_partial_kernel(const float* __restrict__ Kin,
                                    const int*   __restrict__ clen,
                                    const float* __restrict__ V,
                                    float* __restrict__ pm,
                                    float* __restrict__ pl,
                                    float* __restrict__ pctx) {
    const int b     = blockIdx.y;
    const int chunk = blockIdx.x;
    const int tid   = threadIdx.x;
    const int wave  = tid >> 5;
    const int lane  = tid & 31;
    const int len   = clen[b];
    const int sbase = chunk * SCHUNK;

    __shared__ float sm[8];
    __shared__ float sl[8];
    __shared__ __align__(16) float sctx[8 * FEAT_];

    // per-lane slice of v[b]
    const float4* v4 = (const float4*)(V + (size_t)b * FEAT_);
    float4 vv[4];
#pragma unroll
    for (int i = 0; i < 4; ++i) vv[i] = v4[lane + 32 * i];

    float m = -3.0e38f, l = 0.0f;
    float4 acc[4];
#pragma unroll
    for (int i = 0; i < 4; ++i) { acc[i].x = acc[i].y = acc[i].z = acc[i].w = 0.0f; }

    const float4* kb = (const float4*)(Kin + (size_t)b * S_ * FEAT_);

    for (int j = 0; j < SCHUNK / 8; ++j) {
        const int s = sbase + wave + 8 * j;
        if (s >= len) break;  // s monotonically increasing per wave

        // prefetch next row for this wave (lanes stride 64B -> whole 2KB row)
        if (s + 8 < len)
            __builtin_prefetch((const void*)(kb + (size_t)(s + 8) * 128 + lane * 4), 0, 1);

        const float4* krow = kb + (size_t)s * 128;
        float4 kk[4];
#pragma unroll
        for (int i = 0; i < 4; ++i) kk[i] = krow[lane + 32 * i];

        // score = k_row . v  (per-lane partial, then wave32 xor-reduce)
        float d = 0.0f;
#pragma unroll
        for (int i = 0; i < 4; ++i) {
            d = fmaf(kk[i].x, vv[i].x, d);
            d = fmaf(kk[i].y, vv[i].y, d);
            d = fmaf(kk[i].z, vv[i].z, d);
            d = fmaf(kk[i].w, vv[i].w, d);
        }
#pragma unroll
        for (int off = 16; off > 0; off >>= 1)
            d += __shfl_xor(d, off, 32);

        // online softmax update
        const float mnew  = fmaxf(m, d);
        const float scale = __expf(m - mnew);   // underflows to 0 on first hit
        const float p     = __expf(d - mnew);
        l = l * scale + p;
#pragma unroll
        for (int i = 0; i < 4; ++i) {
            acc[i].x = fmaf(p, kk[i].x, acc[i].x * scale);
            acc[i].y = fmaf(p, kk[i].y, acc[i].y * scale);
            acc[i].z = fmaf(p, kk[i].z, acc[i].z * scale);
            acc[i].w = fmaf(p, kk[i].w, acc[i].w * scale);
        }
        m = mnew;
    }

    // stash wave partials in LDS
    if (lane == 0) { sm[wave] = m; sl[wave] = l; }
    float4* dst = (float4*)(sctx + wave * FEAT_);
#pragma unroll
    for (int i = 0; i < 4; ++i) dst[lane + 32 * i] = acc[i];
    __syncthreads();

    // cross-wave combine -> one chunk partial
    float mg = sm[0];
#pragma unroll
    for (int w = 1; w < 8; ++w) mg = fmaxf(mg, sm[w]);

    const size_t pbase = ((size_t)b * NCHUNK + chunk) * FEAT_;
#pragma unroll
    for (int t = 0; t < 2; ++t) {
        const int f = tid + t * 256;
        float sum = 0.0f;
#pragma unroll
        for (int w = 0; w < 8; ++w)
            sum = fmaf(sctx[w * FEAT_ + f], __expf(sm[w] - mg), sum);
        pctx[pbase + f] = sum;
    }
    if (tid == 0) {
        float L = 0.0f;
#pragma unroll
        for (int w = 0; w < 8; ++w)
            L = fmaf(sl[w], __expf(sm[w] - mg), L);
        pl[b * NCHUNK + chunk] = L;
        pm[b * NCHUNK + chunk] = mg;
    }
}

// ---------------------------------------------------------------------------
// Kernel 3: combine NCHUNK partials per batch; len==0 rows -> zero (l==0).
// grid: 128 blocks x 256 threads.
// ---------------------------------------------------------------------------
__global__ void attn_combine_kernel(const float* __restrict__ pm,
                                    const float* __restrict__ pl,
                                    const float* __restrict__ pctx,
                                    float* __restrict__ out) {
    const int b   = blockIdx.x;
    const int tid = threadIdx.x;

    float mg = -3.0e38f;
#pragma unroll
    for (int c = 0; c < NCHUNK; ++c) mg = fmaxf(mg, pm[b * NCHUNK + c]);
    float L = 0.0f;
#pragma unroll
    for (int c = 0; c < NCHUNK; ++c)
        L = fmaf(pl[b * NCHUNK + c], __expf(pm[b * NCHUNK + c] - mg), L);
    const float inv = (L > 0.0f) ? (1.0f / L) : 0.0f;

#pragma unroll
    for (int t = 0; t < 2; ++t) {
        const int f = tid + t * 256;
        float sum = 0.0f;
#pragma unroll
        for (int c = 0; c < NCHUNK; ++c)
            sum = fmaf(pctx[((size_t)b * NCHUNK + c) * FEAT_ + f],
                       __expf(pm[b * NCHUNK + c] - mg), sum);
        out[(size_t)b * FEAT_ + f] = sum * inv;
    }
}

// ---------------------------------------------------------------------------
extern "C" void kernel_launch(void* const* d_in, const int* in_sizes, int n_in,
                              void* d_out, int out_size, void* d_ws, size_t ws_size,
                              hipStream_t stream) {
    const float* input_k = (const float*)d_in[0];   // [B,S,FEAT]
    const float* input_q = (const float*)d_in[1];   // [B,FEAT]
    const int*   clen    = (const int*)d_in[2];     // [B]
    const float* W_K     = (const float*)d_in[3];   // [FEAT,DK]
    const float* W_Q     = (const float*)d_in[4];   // [FEAT,DK]
    float*       out     = (float*)d_out;           // [B,FEAT]

    // workspace layout (floats): q | v | pm | pl | pctx  (~2.3 MB total)
    float* q    = (float*)d_ws;                 // 128*64
    float* v    = q  + B_ * DK_;                // 128*512
    float* pm   = v  + B_ * FEAT_;              // 128*8
    float* pl   = pm + B_ * NCHUNK;             // 128*8
    float* pctx = pl + B_ * NCHUNK;             // 128*8*512

    proj_q_kernel<<<32, 32, 0, stream>>>(input_q, W_Q, q);
    proj_v_kernel<<<256, 32, 0, stream>>>(q, W_K, v);

    dim3 grid2(NCHUNK, B_);
    attn_partial_kernel<<<grid2, 256, 0, stream>>>(input_k, clen, v, pm, pl, pctx);
    attn_combine_kernel<<<B_, 256, 0, stream>>>(pm, pl, pctx, out);
}